// RandomRotation_63127429316811
// MI455X (gfx1250) — compile-verified
//
#include <hip/hip_runtime.h>
#include <hip/hip_bf16.h>
#include <stdint.h>

// ---------------------------------------------------------------------------
// RandomRotation (nearest-neighbor, zero fill) for MI455X / gfx1250.
// Memory-bound gather: ~100 MB traffic -> ~4.3 us floor at 23.3 TB/s.
// No matmul structure -> no WMMA. Input fits in 192 MB L2; output streamed
// with non-temporal stores so it does not pollute the near caches.
// ---------------------------------------------------------------------------

#define BATCH 64
#define HDIM  256
#define WDIM  256
#define CHAN  3

// ----- exact JAX threefry2x32 ------------------------------------------------
__device__ __forceinline__ uint32_t rotl32(uint32_t x, int r) {
    return (x << r) | (x >> (32 - r));
}

__device__ __forceinline__ void threefry2x32(uint32_t k0, uint32_t k1,
                                             uint32_t& x0, uint32_t& x1) {
    const uint32_t ks2 = 0x1BD11BDAu ^ k0 ^ k1;
    const int rotA[4] = {13, 15, 26, 6};
    const int rotB[4] = {17, 29, 16, 24};

    x0 += k0; x1 += k1;
#pragma unroll
    for (int i = 0; i < 4; ++i) { x0 += x1; x1 = rotl32(x1, rotA[i]); x1 ^= x0; }
    x0 += k1; x1 += ks2 + 1u;
#pragma unroll
    for (int i = 0; i < 4; ++i) { x0 += x1; x1 = rotl32(x1, rotB[i]); x1 ^= x0; }
    x0 += ks2; x1 += k0 + 2u;
#pragma unroll
    for (int i = 0; i < 4; ++i) { x0 += x1; x1 = rotl32(x1, rotA[i]); x1 ^= x0; }
    x0 += k0; x1 += k1 + 3u;
#pragma unroll
    for (int i = 0; i < 4; ++i) { x0 += x1; x1 = rotl32(x1, rotB[i]); x1 ^= x0; }
    x0 += k1; x1 += ks2 + 4u;
#pragma unroll
    for (int i = 0; i < 4; ++i) { x0 += x1; x1 = rotl32(x1, rotA[i]); x1 ^= x0; }
    x0 += ks2; x1 += k0 + 5u;
}

// Kernel 1: one wave-pair computes cos/sin per batch into workspace.
// Reproduces jax.random.uniform(jax.random.key(seed), (64,)) semantics:
//   counts = iota(64) split in halves -> hash(ctr=(b, b+32)); b<32 takes x0,
//   b>=32 takes x1 of hash(ctr=(b-32, b)). uniform = bitcast(0x3f800000 |
//   bits>>9) - 1; angle_deg = u*60 - 30.
__global__ void rr_angles_kernel(const int* __restrict__ seed_p,
                                 float* __restrict__ cs /* [BATCH][2] */) {
    int b = threadIdx.x;
    if (b >= BATCH) return;

    int64_t  s64 = (int64_t)seed_p[0];
    uint32_t k0  = (uint32_t)(((uint64_t)s64) >> 32);
    uint32_t k1  = (uint32_t)((uint64_t)s64 & 0xFFFFFFFFull);

    const int half = BATCH / 2; // 32
    uint32_t c0, c1;
    if (b < half) { c0 = (uint32_t)b;          c1 = (uint32_t)(b + half); }
    else          { c0 = (uint32_t)(b - half); c1 = (uint32_t)b; }
    threefry2x32(k0, k1, c0, c1);
    uint32_t bits = (b < half) ? c0 : c1;

    float u   = __uint_as_float(0x3F800000u | (bits >> 9)) - 1.0f; // [0,1)
    float deg = u * 60.0f - 30.0f;
    float ang = deg * 0.017453292519943295f; // pi/180

    cs[2 * b + 0] = cosf(ang);
    cs[2 * b + 1] = sinf(ang);
}

// Kernel 2: one thread per output pixel. Block = one output row (256 thr,
// 8 wave32). blockIdx.x = row, blockIdx.y = batch.
__global__ __launch_bounds__(256)
void rr_apply_kernel(const float* __restrict__ data,
                     const float* __restrict__ cs,
                     float* __restrict__ out) {
    const int x = threadIdx.x;      // column 0..255
    const int y = blockIdx.x;       // row    0..255
    const int b = blockIdx.y;       // batch  0..63

    // Uniform per block -> scalar loads of cos/sin.
    const float c = cs[2 * b + 0];
    const float s = cs[2 * b + 1];

    const float off = 0.5f * (float)(HDIM - 1); // 127.5 (H==W)
    const float fy = (float)y - off;
    const float fx = (float)x - off;

    // newp = R @ (p - off) + off, R = [[c,-s],[s,c]], p = (y, x)
    const float ny = fmaf(c, fy, fmaf(-s, fx, off));
    const float nx = fmaf(s, fy, fmaf( c, fx, off));

    // jnp.round == round-half-to-even == rintf under default RNE mode.
    const int iy = (int)rintf(ny);
    const int ix = (int)rintf(nx);

    const bool oob = (iy < 0) | (iy >= HDIM) | (ix < 0) | (ix >= WDIM);
    const int yc = min(max(iy, 0), HDIM - 1);
    const int xc = min(max(ix, 0), WDIM - 1);

    float v0 = 0.0f, v1 = 0.0f, v2 = 0.0f;
    if (!oob) {
        // Gather: rotated coords of a wave form a nearly contiguous segment;
        // full image (768 KB/batch, 48 MB total) resides in the 192 MB L2.
        const float* __restrict__ src =
            data + ((size_t)b * HDIM + (size_t)yc) * (WDIM * CHAN)
                 + (size_t)xc * CHAN;
        v0 = src[0];
        v1 = src[1];
        v2 = src[2];
    }

    // Streamed output: non-temporal stores (TH=NT) keep the write-once output
    // from displacing the gather working set in near caches.
    float* __restrict__ dst =
        out + ((size_t)b * HDIM + (size_t)y) * (WDIM * CHAN)
            + (size_t)x * CHAN;
    __builtin_nontemporal_store(v0, dst + 0);
    __builtin_nontemporal_store(v1, dst + 1);
    __builtin_nontemporal_store(v2, dst + 2);
}

extern "C" void kernel_launch(void* const* d_in, const int* in_sizes, int n_in,
                              void* d_out, int out_size, void* d_ws, size_t ws_size,
                              hipStream_t stream) {
    const float* data   = (const float*)d_in[0];
    const int*   seed_p = (const int*)d_in[1];
    float*       outp   = (float*)d_out;
    float*       cs     = (float*)d_ws; // BATCH*2 floats = 512 B scratch

    (void)in_sizes; (void)n_in; (void)out_size; (void)ws_size;

    // 1) per-batch angles -> cos/sin in workspace (same-stream ordering
    //    guarantees visibility to the apply kernel).
    rr_angles_kernel<<<1, BATCH, 0, stream>>>(seed_p, cs);

    // 2) rotate-gather: grid (rows, batches), one row per 256-thread block.
    dim3 grid(HDIM, BATCH, 1);
    dim3 block(WDIM, 1, 1);
    rr_apply_kernel<<<grid, block, 0, stream>>>(data, cs, outp);
}